// FeaturePropagation_48421461295500
// MI455X (gfx1250) — compile-verified
//
#include <hip/hip_runtime.h>
#include <hip/hip_bf16.h>
#include <float.h>

// ---------------------------------------------------------------------------
// Problem constants
// ---------------------------------------------------------------------------
#define BB   16
#define NN   4096
#define MM   1024
#define C1   128
#define C2   256
#define INCH 384
#define MLP0 256
#define MLP1 256
#define BN   (BB * NN)          // 65536 rows
#define EPS  1e-5f

typedef float v2f __attribute__((ext_vector_type(2)));
typedef float v8f __attribute__((ext_vector_type(8)));

// ---------------------------------------------------------------------------
// Kernel 1: 3-NN + inverse-distance interpolation, producing the FULL packed
// GEMM1 A operand: featpk[kpair][row][2] with kpair = k/2, row = b*N+n.
//   k in [0,256):   interpolated points2 features
//   k in [256,384): points1 (copied from its natural (c, n) layout)
// Packed-pair layout makes every store here an 8B coalesced write and every
// WMMA A-fragment load in gemm1 a single aligned global_load_b64.
// ---------------------------------------------------------------------------
__global__ void knn_feat_kernel(const float* __restrict__ xyz1,
                                const float* __restrict__ xyz2,
                                const float* __restrict__ points1,
                                const float* __restrict__ points2,
                                float* __restrict__ featpk) {   // [192][BN][2]
    __shared__ float sx[MM];
    __shared__ float sy[MM];
    __shared__ float sz[MM];

    const int tid   = threadIdx.x;
    const int ntile = blockIdx.x % (NN / 256);
    const int b     = blockIdx.x / (NN / 256);
    const int n     = ntile * 256 + tid;

    // stage xyz2[b] (3 x 1024 = 12 KB) into LDS
    const float* x2 = xyz2 + b * 3 * MM;
    for (int i = tid; i < MM; i += 256) {
        sx[i] = x2[0 * MM + i];
        sy[i] = x2[1 * MM + i];
        sz[i] = x2[2 * MM + i];
    }
    __syncthreads();

    const float px = xyz1[b * 3 * NN + 0 * NN + n];
    const float py = xyz1[b * 3 * NN + 1 * NN + n];
    const float pz = xyz1[b * 3 * NN + 2 * NN + n];

    float b0 = FLT_MAX, b1 = FLT_MAX, b2 = FLT_MAX;
    int   i0 = 0, i1 = 0, i2 = 0;
    for (int m = 0; m < MM; ++m) {
        const float dx = px - sx[m];
        const float dy = py - sy[m];
        const float dz = pz - sz[m];
        const float d  = dx * dx + dy * dy + dz * dz;
        if (d < b2) {
            if (d < b1) {
                b2 = b1; i2 = i1;
                if (d < b0) { b1 = b0; i1 = i0; b0 = d; i0 = m; }
                else        { b1 = d;  i1 = m; }
            } else { b2 = d; i2 = m; }
        }
    }

    const float inv0 = 1.0f / fmaxf(b0, 1e-10f);
    const float inv1 = 1.0f / fmaxf(b1, 1e-10f);
    const float inv2 = 1.0f / fmaxf(b2, 1e-10f);
    const float rs   = 1.0f / (inv0 + inv1 + inv2);
    const float w0 = inv0 * rs, w1 = inv1 * rs, w2 = inv2 * rs;

    const int gcol = b * NN + n;

    // interpolated channels: kpair 0..127
    const float* p2 = points2 + b * C2 * MM;
    for (int c = 0; c < C2; c += 2) {
        const float* r0 = p2 + c * MM;
        const float* r1 = r0 + MM;
        v2f v;
        v.x = w0 * r0[i0] + w1 * r0[i1] + w2 * r0[i2];
        v.y = w0 * r1[i0] + w1 * r1[i1] + w2 * r1[i2];
        *(v2f*)(featpk + (size_t)(c >> 1) * (2 * BN) + (size_t)gcol * 2) = v;
    }
    // points1 channels: kpair 128..191 (coalesced along n both read & write)
    const float* p1 = points1 + b * C1 * NN + n;
    for (int j = 0; j < C1; j += 2) {
        v2f v;
        v.x = p1[j * NN];
        v.y = p1[(j + 1) * NN];
        *(v2f*)(featpk + (size_t)(128 + (j >> 1)) * (2 * BN) + (size_t)gcol * 2) = v;
    }
}

// ---------------------------------------------------------------------------
// GEMM1: h1[BN][256] = feat[BN][384] @ W1^T
// Wave computes a 32x32 tile: 2 A-frags x 2 B-frags -> 4 independent
// V_WMMA_F32_16X16X4_F32 per K-step (breaks the accumulator RAW chain,
// 8 FLOP/B cache intensity). Block = 8 waves = 64x128 tile.
// ---------------------------------------------------------------------------
__global__ void gemm1_kernel(const float* __restrict__ featpk,  // [192][BN][2]
                             const float* __restrict__ W1,      // [256][384]
                             float* __restrict__ h1) {          // [BN][256]
    const int wave = threadIdx.x >> 5;       // 8 waves: 2 (rows) x 4 (cols)
    const int lane = threadIdx.x & 31;
    const int wr = wave >> 2, wc = wave & 3;

    const int rowblk = blockIdx.x % (BN / 64);
    const int colblk = blockIdx.x / (BN / 64);
    const int rowbase = rowblk * 64 + wr * 32;
    const int colbase = colblk * 128 + wc * 32;

    const int m    = lane & 15;
    const int half = lane >> 4;              // 0: K=k0,k0+1   1: K=k0+2,k0+3
    const int row0 = rowbase + m;
    const int col0 = colbase + m;

    // A stream base: featpk[(k0/2 + half)][row][2]
    const float* aptr0 = featpk + (size_t)row0 * 2 + (size_t)half * (2 * BN);
    const float* aptr1 = aptr0 + 32;                       // row0 + 16
    const float* wptr0 = W1 + (size_t)col0 * INCH + half * 2;
    const float* wptr1 = wptr0 + 16 * INCH;                // col0 + 16

    v8f acc00 = {}, acc01 = {}, acc10 = {}, acc11 = {};
#pragma unroll 2
    for (int k0 = 0; k0 < INCH; k0 += 4) {
        const size_t aoff = (size_t)k0 * BN;               // (k0/2)*2*BN
        v2f af0 = *(const v2f*)(aptr0 + aoff);
        v2f af1 = *(const v2f*)(aptr1 + aoff);
        v2f bf0 = *(const v2f*)(wptr0 + k0);
        v2f bf1 = *(const v2f*)(wptr1 + k0);
        __builtin_prefetch(aptr0 + aoff + (size_t)8 * BN, 0, 1);
        acc00 = __builtin_amdgcn_wmma_f32_16x16x4_f32(false, af0, false, bf0,
                                                      (short)0, acc00, false, false);
        acc01 = __builtin_amdgcn_wmma_f32_16x16x4_f32(false, af0, false, bf1,
                                                      (short)0, acc01, false, false);
        acc10 = __builtin_amdgcn_wmma_f32_16x16x4_f32(false, af1, false, bf0,
                                                      (short)0, acc10, false, false);
        acc11 = __builtin_amdgcn_wmma_f32_16x16x4_f32(false, af1, false, bf1,
                                                      (short)0, acc11, false, false);
    }

    // D layout: VGPR v holds row (v + 8*half), col (lane&15)
    const int mrow0 = rowbase + (half << 3);
    const int c0 = colbase + m, c1 = colbase + 16 + m;
#pragma unroll
    for (int v = 0; v < 8; ++v) {
        h1[(mrow0 + v) * MLP0 + c0]        = acc00[v];
        h1[(mrow0 + v) * MLP0 + c1]        = acc01[v];
        h1[(mrow0 + 16 + v) * MLP0 + c0]   = acc10[v];
        h1[(mrow0 + 16 + v) * MLP0 + c1]   = acc11[v];
    }
}

// ---------------------------------------------------------------------------
// GEMM2: h2[BN][256] = h1n[BN][256] @ W2^T  (same 32x32 wave tiling)
// ---------------------------------------------------------------------------
__global__ void gemm2_kernel(const float* __restrict__ A,   // [BN][256]
                             const float* __restrict__ W2,  // [256][256]
                             float* __restrict__ h2) {      // [BN][256]
    const int wave = threadIdx.x >> 5;
    const int lane = threadIdx.x & 31;
    const int wr = wave >> 2, wc = wave & 3;

    const int rowblk = blockIdx.x % (BN / 64);
    const int colblk = blockIdx.x / (BN / 64);
    const int rowbase = rowblk * 64 + wr * 32;
    const int colbase = colblk * 128 + wc * 32;

    const int m    = lane & 15;
    const int half = lane >> 4;
    const int row0 = rowbase + m;
    const int col0 = colbase + m;

    const float* aptr0 = A + (size_t)row0 * MLP0 + half * 2;
    const float* aptr1 = aptr0 + 16 * MLP0;
    const float* wptr0 = W2 + (size_t)col0 * MLP0 + half * 2;
    const float* wptr1 = wptr0 + 16 * MLP0;

    v8f acc00 = {}, acc01 = {}, acc10 = {}, acc11 = {};
#pragma unroll 2
    for (int k0 = 0; k0 < MLP0; k0 += 4) {
        v2f af0 = *(const v2f*)(aptr0 + k0);
        v2f af1 = *(const v2f*)(aptr1 + k0);
        v2f bf0 = *(const v2f*)(wptr0 + k0);
        v2f bf1 = *(const v2f*)(wptr1 + k0);
        __builtin_prefetch(aptr0 + k0 + 64, 0, 1);
        acc00 = __builtin_amdgcn_wmma_f32_16x16x4_f32(false, af0, false, bf0,
                                                      (short)0, acc00, false, false);
        acc01 = __builtin_amdgcn_wmma_f32_16x16x4_f32(false, af0, false, bf1,
                                                      (short)0, acc01, false, false);
        acc10 = __builtin_amdgcn_wmma_f32_16x16x4_f32(false, af1, false, bf0,
                                                      (short)0, acc10, false, false);
        acc11 = __builtin_amdgcn_wmma_f32_16x16x4_f32(false, af1, false, bf1,
                                                      (short)0, acc11, false, false);
    }

    const int mrow0 = rowbase + (half << 3);
    const int c0 = colbase + m, c1 = colbase + 16 + m;
#pragma unroll
    for (int v = 0; v < 8; ++v) {
        h2[(mrow0 + v) * MLP1 + c0]        = acc00[v];
        h2[(mrow0 + v) * MLP1 + c1]        = acc01[v];
        h2[(mrow0 + 16 + v) * MLP1 + c0]   = acc10[v];
        h2[(mrow0 + 16 + v) * MLP1 + c1]   = acc11[v];
    }
}

// ---------------------------------------------------------------------------
// BN statistics: per-channel partial sums (deterministic, no atomics)
// ---------------------------------------------------------------------------
__global__ void reduce_stats_kernel(const float* __restrict__ h,   // [BN][256]
                                    float* __restrict__ psum,      // [256][256]
                                    float* __restrict__ psq) {
    const int c  = threadIdx.x;
    const int r0 = blockIdx.x * 256;
    float s = 0.f, q = 0.f;
    for (int r = 0; r < 256; ++r) {
        const float v = h[(size_t)(r0 + r) * 256 + c];
        s += v;
        q += v * v;
    }
    psum[blockIdx.x * 256 + c] = s;
    psq [blockIdx.x * 256 + c] = q;
}

__global__ void finalize_stats_kernel(const float* __restrict__ psum,
                                      const float* __restrict__ psq,
                                      const float* __restrict__ gamma,
                                      const float* __restrict__ beta,
                                      float* __restrict__ scale,
                                      float* __restrict__ bias) {
    const int c = threadIdx.x;
    float s = 0.f, q = 0.f;
    for (int p = 0; p < 256; ++p) {
        s += psum[p * 256 + c];
        q += psq [p * 256 + c];
    }
    const float invc = 1.0f / (float)BN;
    const float mean = s * invc;
    const float var  = q * invc - mean * mean;
    const float sc   = gamma[c] * rsqrtf(var + EPS);
    scale[c] = sc;
    bias[c]  = beta[c] - mean * sc;
}

// ---------------------------------------------------------------------------
// In-place BN + ReLU
// ---------------------------------------------------------------------------
__global__ void bn_relu_kernel(float* __restrict__ h,
                               const float* __restrict__ scale,
                               const float* __restrict__ bias) {
    const int total = BN * 256;
    for (int i = blockIdx.x * blockDim.x + threadIdx.x; i < total;
         i += gridDim.x * blockDim.x) {
        const int c = i & 255;
        h[i] = fmaxf(h[i] * scale[c] + bias[c], 0.0f);
    }
}

// ---------------------------------------------------------------------------
// Final BN + ReLU + transpose to channel-first output (B, 256, N)
// ---------------------------------------------------------------------------
__global__ void bn_transpose_out_kernel(const float* __restrict__ h2,  // [BN][256]
                                        const float* __restrict__ scale,
                                        const float* __restrict__ bias,
                                        float* __restrict__ out) {     // [B][256][N]
    __shared__ float tile[32][33];
    int t = blockIdx.x;
    const int ob = t & 7;   t >>= 3;          // 8 channel tiles
    const int nb = t & 127; t >>= 7;          // 128 n tiles
    const int b  = t;                         // 16 batches
    const int cbase = ob * 32;
    const int nbase = nb * 32;

    const int cc = threadIdx.x & 31;
    const int rr = threadIdx.x >> 5;
#pragma unroll
    for (int it = 0; it < 4; ++it) {
        const int r = rr + it * 8;
        const int c = cbase + cc;
        float v = h2[(size_t)(b * NN + nbase + r) * 256 + c];
        tile[r][cc] = fmaxf(v * scale[c] + bias[c], 0.0f);
    }
    __syncthreads();
#pragma unroll
    for (int it = 0; it < 4; ++it) {
        const int oo = rr + it * 8;
        const int nn = cc;
        out[(size_t)(b * 256 + cbase + oo) * NN + nbase + nn] = tile[nn][oo];
    }
}

// ---------------------------------------------------------------------------
// Host-side launcher
// ---------------------------------------------------------------------------
extern "C" void kernel_launch(void* const* d_in, const int* in_sizes, int n_in,
                              void* d_out, int out_size, void* d_ws, size_t ws_size,
                              hipStream_t stream) {
    const float* xyz1    = (const float*)d_in[0];
    const float* xyz2    = (const float*)d_in[1];
    const float* points1 = (const float*)d_in[2];
    const float* points2 = (const float*)d_in[3];
    const float* W1      = (const float*)d_in[4];
    const float* gamma1  = (const float*)d_in[5];
    const float* beta1   = (const float*)d_in[6];
    const float* W2      = (const float*)d_in[7];
    const float* gamma2  = (const float*)d_in[8];
    const float* beta2   = (const float*)d_in[9];
    float* out = (float*)d_out;

    // workspace layout (floats)
    float* ws     = (float*)d_ws;
    float* featpk = ws;                              // [192][BN][2] packed; later h2
    float* h1     = featpk + (size_t)INCH * BN;      // [BN][256]
    float* psum   = h1   + (size_t)BN * 256;         // [256][256]
    float* psq    = psum + 256 * 256;                // [256][256]
    float* scale1 = psq  + 256 * 256;
    float* bias1  = scale1 + 256;
    float* scale2 = bias1  + 256;
    float* bias2  = scale2 + 256;
    float* h2     = featpk;                          // reuse big buffer

    // 1. 3-NN + interpolation + feature packing -> featpk
    knn_feat_kernel<<<BB * (NN / 256), 256, 0, stream>>>(xyz1, xyz2, points1,
                                                         points2, featpk);

    // 2. GEMM1 (fp32 WMMA, 32x32 wave tiles) -> h1
    gemm1_kernel<<<(BN / 64) * (MLP0 / 128), 256, 0, stream>>>(featpk, W1, h1);

    // 3. BN1 stats (deterministic two-pass) + BN+ReLU in place
    reduce_stats_kernel<<<BN / 256, 256, 0, stream>>>(h1, psum, psq);
    finalize_stats_kernel<<<1, 256, 0, stream>>>(psum, psq, gamma1, beta1, scale1, bias1);
    bn_relu_kernel<<<8192, 256, 0, stream>>>(h1, scale1, bias1);

    // 4. GEMM2 -> h2 (reuses featpk buffer)
    gemm2_kernel<<<(BN / 64) * (MLP1 / 128), 256, 0, stream>>>(h1, W2, h2);

    // 5. BN2 stats + fused BN+ReLU+transpose to (B, 256, N)
    reduce_stats_kernel<<<BN / 256, 256, 0, stream>>>(h2, psum, psq);
    finalize_stats_kernel<<<1, 256, 0, stream>>>(psum, psq, gamma2, beta2, scale2, bias2);
    bn_transpose_out_kernel<<<BB * (NN / 32) * (256 / 32), 256, 0, stream>>>(
        h2, scale2, bias2, out);
}